// SNRContrastiveLoss_80951543595538
// MI455X (gfx1250) — compile-verified
//
#include <hip/hip_runtime.h>
#include <hip/hip_bf16.h>

typedef __attribute__((ext_vector_type(2))) float v2f;
typedef __attribute__((ext_vector_type(8))) float v8f;

#define N_ROWS 8192
#define DIM 256
#define CORR (256.0f / 255.0f)
#define DINV (1.0f / 256.0f)
#define POS_MARGIN 0.01f
#define NEG_MARGIN 0.2f
#define REG_WEIGHT 0.1f

// Workspace layout (float units):
//   [0, N*D)          normalized embeddings e (row-major fp32)
//   [N*D, +N)         m[i]   (row mean of e)
//   [+N, +N)          iv[i]  (1 / row_var[i])
//   accumulators: acc[0]=pos_sum, acc[1]=neg_sum, acc[2]=reg_sum,
//                 acc[3]=pos_cnt (u32), acc[4]=neg_cnt (u32)

__global__ void snr_init_kernel(float* acc) {
    if (threadIdx.x == 0) {
        acc[0] = 0.0f; acc[1] = 0.0f; acc[2] = 0.0f;
        ((unsigned int*)acc)[3] = 0u;
        ((unsigned int*)acc)[4] = 0u;
    }
}

// One wave (32 lanes) per row: 8 floats per lane.
__global__ __launch_bounds__(256) void snr_rowstats_kernel(
    const float* __restrict__ x, float* __restrict__ e,
    float* __restrict__ m, float* __restrict__ iv, float* __restrict__ acc) {
    const int row  = blockIdx.x * 8 + (threadIdx.x >> 5);
    const int lane = threadIdx.x & 31;
    const float4* xr = (const float4*)(x + (size_t)row * DIM);
    float4 a = xr[lane];
    float4 b = xr[lane + 32];

    float sum = a.x + a.y + a.z + a.w + b.x + b.y + b.z + b.w;
    float sq  = a.x*a.x + a.y*a.y + a.z*a.z + a.w*a.w
              + b.x*b.x + b.y*b.y + b.z*b.z + b.w*b.w;
    #pragma unroll
    for (int o = 16; o > 0; o >>= 1) {
        sum += __shfl_xor(sum, o, 32);
        sq  += __shfl_xor(sq,  o, 32);
    }
    const float rn = 1.0f / sqrtf(sq);   // 1/||row||

    float4* er = (float4*)(e + (size_t)row * DIM);
    a.x *= rn; a.y *= rn; a.z *= rn; a.w *= rn;
    b.x *= rn; b.y *= rn; b.z *= rn; b.w *= rn;
    er[lane]      = a;
    er[lane + 32] = b;

    if (lane == 0) {
        const float mr = sum * rn * DINV;                 // mean of normalized row
        m[row]  = mr;
        iv[row] = 1.0f / (CORR * (DINV - mr * mr));       // 1 / row_var
        atomicAdd(acc + 2, fabsf(sum) * rn);              // |sum(e_row)| = |m|*D
    }
}

// 128 threads = 4 waves; each wave owns a 64x64 tile of the Gram matrix via
// 4x4 register blocking of 16x16 V_WMMA_F32_16X16X4_F32 tiles (fp32-exact).
__global__ __launch_bounds__(128) void snr_pairwise_kernel(
    const float* __restrict__ e, const float* __restrict__ m,
    const float* __restrict__ iv, const int* __restrict__ labels,
    float* __restrict__ acc) {
    __shared__ float s_mi[128], s_iv[128], s_mj[128];
    __shared__ int   s_li[128], s_lj[128];

    const int t  = threadIdx.x;
    const int i0 = blockIdx.y * 128;
    const int j0 = blockIdx.x * 128;
    s_mi[t] = m[i0 + t];
    s_iv[t] = iv[i0 + t];
    s_li[t] = labels[i0 + t];
    s_mj[t] = m[j0 + t];
    s_lj[t] = labels[j0 + t];
    __syncthreads();

    const int lane = t & 31;
    const int wave = t >> 5;
    const int lm = lane & 15;        // matrix row/col within 16
    const int hi = lane >> 4;        // K half-select for fp32 fragments
    const int wi = (wave >> 1) * 64; // wave's i-offset inside 128-block
    const int wj = (wave & 1) * 64;  // wave's j-offset inside 128-block

    const float* aBase = e + (size_t)(i0 + wi + lm) * DIM + 2 * hi;
    const float* bBase = e + (size_t)(j0 + wj + lm) * DIM + 2 * hi;

    v8f accT[4][4];
    #pragma unroll
    for (int r = 0; r < 4; ++r)
        #pragma unroll
        for (int c = 0; c < 4; ++c)
            accT[r][c] = (v8f){0.f, 0.f, 0.f, 0.f, 0.f, 0.f, 0.f, 0.f};

    for (int k = 0; k < DIM; k += 4) {
        v2f a[4], b[4];
        #pragma unroll
        for (int r = 0; r < 4; ++r)
            a[r] = *(const v2f*)(aBase + (size_t)r * 16 * DIM + k);
        #pragma unroll
        for (int c = 0; c < 4; ++c)
            b[c] = *(const v2f*)(bBase + (size_t)c * 16 * DIM + k);
        #pragma unroll
        for (int r = 0; r < 4; ++r)
            #pragma unroll
            for (int c = 0; c < 4; ++c)
                accT[r][c] = __builtin_amdgcn_wmma_f32_16x16x4_f32(
                    false, a[r], false, b[c], (short)0, accT[r][c], false, false);
    }

    // Fused epilogue: distmat + margins + masked sums/counts.
    float psum = 0.f, nsum = 0.f;
    unsigned int pcnt = 0u, ncnt = 0u;
    #pragma unroll
    for (int r = 0; r < 4; ++r) {
        #pragma unroll
        for (int c = 0; c < 4; ++c) {
            #pragma unroll
            for (int v = 0; v < 8; ++v) {
                const int il = wi + r * 16 + 8 * hi + v;  // C/D layout: M = v + 8*hi
                const int jl = wj + c * 16 + lm;          //             N = lane&15
                const float dot  = accT[r][c][v];
                const float dm   = s_mi[il] - s_mj[jl];
                const float dist = CORR * (2.0f * DINV * (1.0f - dot) - dm * dm)
                                   * s_iv[il];
                if (s_li[il] == s_lj[jl]) {
                    if (i0 + il != j0 + jl) {
                        const float p = dist - POS_MARGIN;
                        if (p > 0.f) { psum += p; pcnt++; }
                    }
                } else {
                    const float q = NEG_MARGIN - dist;
                    if (q > 0.f) { nsum += q; ncnt++; }
                }
            }
        }
    }

    #pragma unroll
    for (int o = 16; o > 0; o >>= 1) {
        psum += __shfl_down(psum, o, 32);
        nsum += __shfl_down(nsum, o, 32);
        pcnt += __shfl_down(pcnt, o, 32);
        ncnt += __shfl_down(ncnt, o, 32);
    }
    if (lane == 0) {
        atomicAdd(acc + 0, psum);
        atomicAdd(acc + 1, nsum);
        atomicAdd(((unsigned int*)acc) + 3, pcnt);
        atomicAdd(((unsigned int*)acc) + 4, ncnt);
    }
}

__global__ void snr_finalize_kernel(const float* __restrict__ acc,
                                    float* __restrict__ out) {
    if (threadIdx.x == 0) {
        const float psum = acc[0];
        const float nsum = acc[1];
        const float reg  = acc[2];
        const float pcnt = (float)((const unsigned int*)acc)[3];
        const float ncnt = (float)((const unsigned int*)acc)[4];
        out[0] = psum / (pcnt + 1e-12f)
               + nsum / (ncnt + 1e-12f)
               + REG_WEIGHT * reg / (float)N_ROWS;
    }
}

extern "C" void kernel_launch(void* const* d_in, const int* in_sizes, int n_in,
                              void* d_out, int out_size, void* d_ws, size_t ws_size,
                              hipStream_t stream) {
    const float* x      = (const float*)d_in[0];
    const int*   labels = (const int*)d_in[1];

    float* ws = (float*)d_ws;
    float* e   = ws;
    float* m   = ws + (size_t)N_ROWS * DIM;
    float* iv  = m + N_ROWS;
    float* acc = iv + N_ROWS;

    snr_init_kernel<<<1, 32, 0, stream>>>(acc);
    snr_rowstats_kernel<<<N_ROWS / 8, 256, 0, stream>>>(x, e, m, iv, acc);
    dim3 grid(N_ROWS / 128, N_ROWS / 128);
    snr_pairwise_kernel<<<grid, 128, 0, stream>>>(e, m, iv, labels, acc);
    snr_finalize_kernel<<<1, 32, 0, stream>>>(acc, (float*)d_out);
}